// ParallelGNN_22376779612623
// MI455X (gfx1250) — compile-verified
//
#include <hip/hip_runtime.h>
#include <hip/hip_bf16.h>
#include <math.h>

typedef float v2f __attribute__((ext_vector_type(2)));
typedef float v8f __attribute__((ext_vector_type(8)));

#define NN 50000      // nodes
#define NE 800000     // edges
#define NG 512        // graphs

// ---------- ordered-key float <-> uint for atomic max ----------
__device__ __forceinline__ unsigned f2ord(float f) {
  unsigned u = __float_as_uint(f);
  return (u & 0x80000000u) ? ~u : (u | 0x80000000u);
}
__device__ __forceinline__ float ord2f(unsigned k) {
  unsigned u = (k & 0x80000000u) ? (k ^ 0x80000000u) : ~k;
  return __uint_as_float(u);
}

// ---------- generic fill / copy ----------
__global__ void k_fill0(float* __restrict__ p, int n) {
  int i = blockIdx.x * blockDim.x + threadIdx.x;
  if (i < n) p[i] = 0.0f;
}
__global__ void k_copy4(float4* __restrict__ d, const float4* __restrict__ s, int n4) {
  int i = blockIdx.x * blockDim.x + threadIdx.x;
  if (i < n4) d[i] = s[i];
}

// ---------- WMMA fp32 GEMM: Y[N,M] = act(A[N,K] @ W[K,M] + bias) ----------
// Compile-time K/M/ACT; K-loop fully unrolled. W is staged into LDS
// COLUMN-MAJOR with padded stride (K+4) via CDNA5 async global->LDS copies
// (ASYNCcnt): the B-fragment {W[k][c],W[k+1][c]} is then 8 contiguous aligned
// bytes -> one ds_load_b64 straight into the even-aligned v2f WMMA operand
// (no register shuffles), and the pad keeps columns on distinct bank groups.
// One wave computes a 16x64 strip. N%16==0, K%4==0, M%64==0.
// Fragment layouts per CDNA5 ISA 7.12.2:
//   A 16x4  : lane<16 row=lane K={kk,kk+1}; lane>=16 row=lane-16 K={kk+2,kk+3}
//   B 4x16  : lane<16 col=lane K rows {kk,kk+1}; lane>=16 col=lane-16 K rows {kk+2,kk+3}
//   C/D     : reg r -> row r (lanes 0-15) / row r+8 (lanes 16-31), col = lane%16
template <int K, int M, int ACT>
__global__ void k_gemm(const float* __restrict__ A, const float* __restrict__ W,
                       const float* __restrict__ bias, float* __restrict__ Y, int N) {
  constexpr int LSTRIDE = K + 4;                 // pad -> conflict-free banks
  __shared__ __align__(16) float wlds[LSTRIDE * M];

  // ---- stage W[K,M] row-major global -> column-major LDS, async b32/lane ----
  {
    unsigned lbase = (unsigned)(size_t)(void*)wlds;   // flat addr [31:0] == LDS offset
#pragma unroll 1
    for (int i = (int)threadIdx.x; i < K * M; i += 256) {
      int k = i / M, col = i - k * M;                 // i is contiguous in global W
      unsigned laddr = lbase + (unsigned)((col * LSTRIDE + k) * 4);
      const char* ga = (const char*)(W + i);
      asm volatile("global_load_async_to_lds_b32 %0, %1, off"
                   :: "v"(laddr), "v"(ga) : "memory");
    }
    asm volatile("s_wait_asynccnt 0x0" ::: "memory");
  }
  __syncthreads();

  int wave = blockIdx.x * (blockDim.x >> 5) + (threadIdx.x >> 5);
  constexpr int tilesN = M >> 6;
  int total = (N >> 4) * tilesN;
  if (wave >= total) return;                 // uniform per wave: EXEC stays all-ones
  int rowBase = (wave / tilesN) << 4;
  int colBase = (wave % tilesN) << 6;
  int l = threadIdx.x & 31;
  int half = l >> 4, lm = l & 15;

  const float* Ar = A + (size_t)(rowBase + lm) * K + (half << 1);
  const float* Bc = wlds + (size_t)(colBase + lm) * LSTRIDE + (half << 1);

  v8f acc[4] = {};
#pragma unroll
  for (int kk = 0; kk < K; kk += 4) {
    v2f a = *(const v2f*)(Ar + kk);                       // global_load_b64
#pragma unroll
    for (int tt = 0; tt < 4; ++tt) {
      v2f b = *(const v2f*)(Bc + tt * 16 * LSTRIDE + kk); // ds_load_b64, imm offset
      acc[tt] = __builtin_amdgcn_wmma_f32_16x16x4_f32(
          false, a, false, b, (short)0, acc[tt], false, false);
    }
  }

  float bv[4];
#pragma unroll
  for (int tt = 0; tt < 4; ++tt) bv[tt] = bias[colBase + tt * 16 + lm];
#pragma unroll
  for (int r = 0; r < 8; ++r) {
    size_t row = (size_t)(rowBase + r + (half << 3));
    float* yp = Y + row * (size_t)M + colBase + lm;
#pragma unroll
    for (int tt = 0; tt < 4; ++tt) {
      float vv = acc[tt][r] + bv[tt];
      if (ACT) vv = fmaxf(vv, 0.0f);
      yp[tt * 16] = vv;
    }
  }
}

// ---------- GIN edge scatter: z[dst] += h[src] (64 ch, 16 threads/edge) ----------
__global__ void k_gin_scatter(const float* __restrict__ hin, float* __restrict__ z,
                              const int* __restrict__ ei) {
  int tid = blockIdx.x * blockDim.x + threadIdx.x;
  int e = tid >> 4;
  if (e >= NE) return;
  int c = (tid & 15) << 2;
  int s = ei[e], d = ei[NE + e];
  const float4 hv = *(const float4*)(hin + (size_t)s * 64 + c);
  float* zp = z + (size_t)d * 64 + c;
  atomicAdd(zp + 0, hv.x); atomicAdd(zp + 1, hv.y);
  atomicAdd(zp + 2, hv.z); atomicAdd(zp + 3, hv.w);
}

// ---------- graph pooling: pool[batch[n]] += h[n], optional node count ----------
__global__ void k_pool(const float* __restrict__ h, const int* __restrict__ batch,
                       float* __restrict__ pool, float* __restrict__ cnt, int F) {
  int tid = blockIdx.x * blockDim.x + threadIdx.x;
  int chunks = F >> 2;
  int n = tid / chunks;
  if (n >= NN) return;
  int c = (tid - n * chunks) << 2;
  int b = batch[n];
  const float4 hv = *(const float4*)(h + (size_t)n * F + c);
  float* pp = pool + (size_t)b * F + c;
  atomicAdd(pp + 0, hv.x); atomicAdd(pp + 1, hv.y);
  atomicAdd(pp + 2, hv.z); atomicAdd(pp + 3, hv.w);
  if (cnt != nullptr && c == 0) atomicAdd(cnt + b, 1.0f);
}

// ---------- GT: per-edge alpha + segment max (wave per edge) ----------
__global__ void k_gt_alpha(const float* __restrict__ q, const float* __restrict__ kmat,
                           const float* __restrict__ eattr, const float* __restrict__ We,
                           const int* __restrict__ ei, float* __restrict__ alpha,
                           unsigned* __restrict__ mkey) {
  int e = blockIdx.x * 8 + (threadIdx.x >> 5);
  if (e >= NE) return;
  int l = threadIdx.x & 31;
  int half = l >> 4, lm = l & 15;
  int s = ei[e], d = ei[NE + e];
  float eav = eattr[(size_t)e * 16 + lm];
  int c0 = l << 2;
  float ec0 = 0.f, ec1 = 0.f, ec2 = 0.f, ec3 = 0.f;
#pragma unroll
  for (int dd = 0; dd < 16; ++dd) {
    float ad = __shfl(eav, dd, 32);
    const float* wr = We + dd * 128 + c0;
    ec0 += ad * wr[0]; ec1 += ad * wr[1]; ec2 += ad * wr[2]; ec3 += ad * wr[3];
  }
  const float* kp = kmat + (size_t)s * 128 + c0;
  const float* qp = q + (size_t)d * 128 + c0;
  float p = qp[0] * (kp[0] + ec0) + qp[1] * (kp[1] + ec1) +
            qp[2] * (kp[2] + ec2) + qp[3] * (kp[3] + ec3);
  p += __shfl_xor(p, 1, 32); p += __shfl_xor(p, 2, 32);
  p += __shfl_xor(p, 4, 32); p += __shfl_xor(p, 8, 32);  // 16-lane (per-head) reduce
  if (lm == 0) {
    float a = p * 0.125f;  // 1/sqrt(64)
    alpha[(size_t)e * 2 + half] = a;
    atomicMax(mkey + (size_t)d * 2 + half, f2ord(a));
  }
}

// ---------- GT: ex = exp(alpha - m[dst]); s[dst] += ex (thread per edge-head) ----------
__global__ void k_gt_expsum(const int* __restrict__ ei, float* __restrict__ alpha,
                            const unsigned* __restrict__ mkey, float* __restrict__ s) {
  int tid = blockIdx.x * blockDim.x + threadIdx.x;
  if (tid >= NE * 2) return;
  int e = tid >> 1, h = tid & 1;
  int d = ei[NE + e];
  float mv = ord2f(mkey[(size_t)d * 2 + h]);
  float ex = __expf(alpha[(size_t)e * 2 + h] - mv);
  alpha[(size_t)e * 2 + h] = ex;
  atomicAdd(s + (size_t)d * 2 + h, ex);
}

// ---------- GT: out[dst] += (v[src]+e)*a (wave per edge) ----------
__global__ void k_gt_msg(const float* __restrict__ vmat, const float* __restrict__ eattr,
                         const float* __restrict__ We, const int* __restrict__ ei,
                         const float* __restrict__ alpha, const float* __restrict__ s,
                         float* __restrict__ outb) {
  int e = blockIdx.x * 8 + (threadIdx.x >> 5);
  if (e >= NE) return;
  int l = threadIdx.x & 31;
  int half = l >> 4, lm = l & 15;
  int sn = ei[e], d = ei[NE + e];
  float eav = eattr[(size_t)e * 16 + lm];
  int c0 = l << 2;
  float ec0 = 0.f, ec1 = 0.f, ec2 = 0.f, ec3 = 0.f;
#pragma unroll
  for (int dd = 0; dd < 16; ++dd) {
    float ad = __shfl(eav, dd, 32);
    const float* wr = We + dd * 128 + c0;
    ec0 += ad * wr[0]; ec1 += ad * wr[1]; ec2 += ad * wr[2]; ec3 += ad * wr[3];
  }
  float a = alpha[(size_t)e * 2 + half] / (s[(size_t)d * 2 + half] + 1e-16f);
  const float* vp = vmat + (size_t)sn * 128 + c0;
  float* op = outb + (size_t)d * 128 + c0;
  atomicAdd(op + 0, (vp[0] + ec0) * a);
  atomicAdd(op + 1, (vp[1] + ec1) * a);
  atomicAdd(op + 2, (vp[2] + ec2) * a);
  atomicAdd(op + 3, (vp[3] + ec3) * a);
}

// ---------- GT: beta gate, g = beta*skip + (1-beta)*out (wave per node) ----------
__global__ void k_gt_beta(const float* __restrict__ outb, const float* __restrict__ skip,
                          const float* __restrict__ Wb, float* __restrict__ g) {
  int n = blockIdx.x * 8 + (threadIdx.x >> 5);
  if (n >= NN) return;
  int l = threadIdx.x & 31;
  int c0 = l << 2;
  const float* op = outb + (size_t)n * 128 + c0;
  const float* sp = skip + (size_t)n * 128 + c0;
  float o0 = op[0], o1 = op[1], o2 = op[2], o3 = op[3];
  float s0 = sp[0], s1 = sp[1], s2 = sp[2], s3 = sp[3];
  float p = o0 * Wb[c0] + o1 * Wb[c0 + 1] + o2 * Wb[c0 + 2] + o3 * Wb[c0 + 3]
          + s0 * Wb[128 + c0] + s1 * Wb[128 + c0 + 1] + s2 * Wb[128 + c0 + 2] + s3 * Wb[128 + c0 + 3]
          + (o0 - s0) * Wb[256 + c0] + (o1 - s1) * Wb[256 + c0 + 1]
          + (o2 - s2) * Wb[256 + c0 + 2] + (o3 - s3) * Wb[256 + c0 + 3];
  p += __shfl_xor(p, 1, 32); p += __shfl_xor(p, 2, 32); p += __shfl_xor(p, 4, 32);
  p += __shfl_xor(p, 8, 32); p += __shfl_xor(p, 16, 32);
  float beta = 1.0f / (1.0f + __expf(-p));
  float* gp = g + (size_t)n * 128 + c0;
  gp[0] = beta * s0 + (1.0f - beta) * o0;
  gp[1] = beta * s1 + (1.0f - beta) * o1;
  gp[2] = beta * s2 + (1.0f - beta) * o2;
  gp[3] = beta * s3 + (1.0f - beta) * o3;
}

// ---------- readout helpers ----------
__global__ void k_zc(const float* __restrict__ ginp, const float* __restrict__ gtp,
                     const float* __restrict__ cnt, float* __restrict__ zc) {
  int tid = blockIdx.x * blockDim.x + threadIdx.x;
  if (tid >= NG * 192) return;
  int gg = tid / 192, c = tid - gg * 192;
  float v = (c < 64) ? ginp[gg * 64 + c]
                     : gtp[gg * 128 + (c - 64)] / fmaxf(cnt[gg], 1.0f);
  zc[tid] = v;
}
__global__ void k_final(const float* __restrict__ zh, const float* __restrict__ W2,
                        const float* __restrict__ b2, float* __restrict__ out) {
  int gg = blockIdx.x;
  int l = threadIdx.x;
  float p = zh[gg * 64 + l] * W2[l] + zh[gg * 64 + 32 + l] * W2[32 + l];
  p += __shfl_xor(p, 1, 32); p += __shfl_xor(p, 2, 32); p += __shfl_xor(p, 4, 32);
  p += __shfl_xor(p, 8, 32); p += __shfl_xor(p, 16, 32);
  if (l == 0) out[gg] = p + b2[0];
}

extern "C" void kernel_launch(void* const* d_in, const int* in_sizes, int n_in,
                              void* d_out, int out_size, void* d_ws, size_t ws_size,
                              hipStream_t stream) {
  (void)in_sizes; (void)n_in; (void)out_size; (void)ws_size;

  const float* x     = (const float*)d_in[0];   // [50000,64]
  const float* eattr = (const float*)d_in[1];   // [800000,16]
  const int*   ei    = (const int*)d_in[2];     // [2,800000]
  const int*   batch = (const int*)d_in[3];     // [50000]

  const float *gW1[3], *gb1[3], *gW2[3], *gb2[3];
  for (int i = 0; i < 3; ++i) {
    gW1[i] = (const float*)d_in[4 + i * 4 + 0];
    gb1[i] = (const float*)d_in[4 + i * 4 + 1];
    gW2[i] = (const float*)d_in[4 + i * 4 + 2];
    gb2[i] = (const float*)d_in[4 + i * 4 + 3];
  }
  const float *Wq[2], *bq[2], *Wk[2], *bk[2], *Wv[2], *bv[2], *We[2], *Wsk[2], *bsk[2], *Wb[2];
  for (int j = 0; j < 2; ++j) {
    int b = 16 + j * 10;
    Wq[j]  = (const float*)d_in[b + 0]; bq[j]  = (const float*)d_in[b + 1];
    Wk[j]  = (const float*)d_in[b + 2]; bk[j]  = (const float*)d_in[b + 3];
    Wv[j]  = (const float*)d_in[b + 4]; bv[j]  = (const float*)d_in[b + 5];
    We[j]  = (const float*)d_in[b + 6];
    Wsk[j] = (const float*)d_in[b + 7]; bsk[j] = (const float*)d_in[b + 8];
    Wb[j]  = (const float*)d_in[b + 9];
  }
  const float* roW1 = (const float*)d_in[36];
  const float* rob1 = (const float*)d_in[37];
  const float* roW2 = (const float*)d_in[38];
  const float* rob2 = (const float*)d_in[39];
  float* out = (float*)d_out;

  // ---- workspace bump allocator ----
  char* wp = (char*)d_ws;
  auto alloc = [&](size_t nbytes) -> void* {
    void* p = (void*)wp;
    wp += (nbytes + 255) & ~(size_t)255;
    return p;
  };
  float* h    = (float*)alloc((size_t)NN * 64 * 4);
  float* z    = (float*)alloc((size_t)NN * 64 * 4);
  float* t    = (float*)alloc((size_t)NN * 64 * 4);
  float* g    = (float*)alloc((size_t)NN * 128 * 4);
  float* q    = (float*)alloc((size_t)NN * 128 * 4);
  float* kbuf = (float*)alloc((size_t)NN * 128 * 4);
  float* vbuf = (float*)alloc((size_t)NN * 128 * 4);
  float* sk   = (float*)alloc((size_t)NN * 128 * 4);
  float* ob   = (float*)alloc((size_t)NN * 128 * 4);
  float* alp  = (float*)alloc((size_t)NE * 2 * 4);
  unsigned* mk = (unsigned*)alloc((size_t)NN * 2 * 4);
  float* ssum = (float*)alloc((size_t)NN * 2 * 4);
  float* ginp = (float*)alloc((size_t)NG * 64 * 4);
  float* gtp  = (float*)alloc((size_t)NG * 128 * 4);
  float* cnt  = (float*)alloc((size_t)NG * 4);
  float* zc   = (float*)alloc((size_t)NG * 192 * 4);
  float* zh   = (float*)alloc((size_t)NG * 64 * 4);

  auto blocksFor = [](int N, int M) { return (((N >> 4) * (M >> 6)) + 7) / 8; };
  auto fill = [&](float* p, int n) {
    k_fill0<<<(n + 255) / 256, 256, 0, stream>>>(p, n);
  };

  // ==== GIN stack (K=64, M=64, relu) ====
  for (int i = 0; i < 3; ++i) {
    const float* in = (i == 0) ? x : h;
    k_copy4<<<(NN * 16 + 255) / 256, 256, 0, stream>>>((float4*)z, (const float4*)in, NN * 16);
    k_gin_scatter<<<(NE * 16 + 255) / 256, 256, 0, stream>>>(in, z, ei);  // z = in + agg
    k_gemm<64, 64, 1><<<blocksFor(NN, 64), 256, 0, stream>>>(z, gW1[i], gb1[i], t, NN);
    k_gemm<64, 64, 1><<<blocksFor(NN, 64), 256, 0, stream>>>(t, gW2[i], gb2[i], h, NN);
  }
  fill(ginp, NG * 64);
  k_pool<<<(NN * 16 + 255) / 256, 256, 0, stream>>>(h, batch, ginp, nullptr, 64);

  // ==== TransformerConv stack ====
  for (int j = 0; j < 2; ++j) {
    int gb = blocksFor(NN, 128);
    if (j == 0) {
      k_gemm<64, 128, 0><<<gb, 256, 0, stream>>>(x, Wq[j],  bq[j],  q,    NN);
      k_gemm<64, 128, 0><<<gb, 256, 0, stream>>>(x, Wk[j],  bk[j],  kbuf, NN);
      k_gemm<64, 128, 0><<<gb, 256, 0, stream>>>(x, Wv[j],  bv[j],  vbuf, NN);
      k_gemm<64, 128, 0><<<gb, 256, 0, stream>>>(x, Wsk[j], bsk[j], sk,   NN);
    } else {
      k_gemm<128, 128, 0><<<gb, 256, 0, stream>>>(g, Wq[j],  bq[j],  q,    NN);
      k_gemm<128, 128, 0><<<gb, 256, 0, stream>>>(g, Wk[j],  bk[j],  kbuf, NN);
      k_gemm<128, 128, 0><<<gb, 256, 0, stream>>>(g, Wv[j],  bv[j],  vbuf, NN);
      k_gemm<128, 128, 0><<<gb, 256, 0, stream>>>(g, Wsk[j], bsk[j], sk,   NN);
    }
    fill((float*)mk, NN * 2);   // ordered key 0 == minimum
    fill(ssum, NN * 2);
    fill(ob, NN * 128);
    k_gt_alpha<<<(NE + 7) / 8, 256, 0, stream>>>(q, kbuf, eattr, We[j], ei, alp, mk);
    k_gt_expsum<<<(NE * 2 + 255) / 256, 256, 0, stream>>>(ei, alp, mk, ssum);
    k_gt_msg<<<(NE + 7) / 8, 256, 0, stream>>>(vbuf, eattr, We[j], ei, alp, ssum, ob);
    k_gt_beta<<<(NN + 7) / 8, 256, 0, stream>>>(ob, sk, Wb[j], g);
  }
  fill(gtp, NG * 128);
  fill(cnt, NG);
  k_pool<<<(NN * 32 + 255) / 256, 256, 0, stream>>>(g, batch, gtp, cnt, 128);

  // ==== readout (K=192, M=64, relu) ====
  k_zc<<<(NG * 192 + 255) / 256, 256, 0, stream>>>(ginp, gtp, cnt, zc);
  k_gemm<192, 64, 1><<<blocksFor(NG, 64), 256, 0, stream>>>(zc, roW1, rob1, zh, NG);
  k_final<<<NG, 32, 0, stream>>>(zh, roW2, rob2, out);
}